// BlurredPhonemeEmbedding_19172734009775
// MI455X (gfx1250) — compile-verified
//
#include <hip/hip_runtime.h>
#include <stdint.h>

#define T_LEN    8192
#define D_DIM    64
#define CHUNK    512
#define HALO     512
#define NTHREADS 256

typedef unsigned int u32;
typedef u32   u32x4 __attribute__((ext_vector_type(4)));
typedef int   i32x4 __attribute__((ext_vector_type(4)));
typedef int   i32x8 __attribute__((ext_vector_type(8)));
typedef float f32x4 __attribute__((ext_vector_type(4)));

__global__ __launch_bounds__(NTHREADS)
void blur_embed_kernel(const int* __restrict__ ids,
                       const float* __restrict__ table,
                       float* __restrict__ out) {
  __shared__ int   sid[CHUNK + 2 * HALO];   // 6 KB ids slice (chunk + halo)
  __shared__ float sw [CHUNK];
  __shared__ int   snb[CHUNK];

  const int chunksPerRow = T_LEN / CHUNK;
  const int b   = blockIdx.x / chunksPerRow;
  const int cb  = (blockIdx.x % chunksPerRow) * CHUNK;
  const int lo  = max(0, cb - HALO);
  const int hi  = min(T_LEN, cb + CHUNK + HALO);
  const int n   = hi - lo;
  const int off = cb - lo;

  // ---- Phase 1: TDM async DMA of ids[b, lo:hi) into LDS --------------------
  // TDM ignores EXEC, so gate with a *scalar* branch (readfirstlane -> SCC
  // branch) so exactly one wave issues the descriptor and waits on it.
  if (__builtin_amdgcn_readfirstlane((int)threadIdx.x) == 0) {
    uint64_t g = (uint64_t)(uintptr_t)(ids + (size_t)b * T_LEN + lo);
    u32 ldsa = (u32)(uintptr_t)(&sid[0]);            // low 32 bits = LDS byte offset
    u32x4 d0;
    d0[0] = 1u;                                      // count=1, valid, no gather
    d0[1] = ldsa;                                    // lds_addr
    d0[2] = (u32)g;                                  // global_addr[31:0]
    d0[3] = (u32)(g >> 32) | (2u << 30);             // global_addr[56:32] | type=2
    i32x8 d1;
    d1[0] = (int)(2u << 16);                         // data_size=4B, wg_mask=0
    d1[1] = (int)((u32)n << 16);                     // tensor_dim0[15:0] = n
    d1[2] = (int)(1u << 16);                         // tensor_dim0 hi=0, tensor_dim1=1
    d1[3] = (int)((u32)n << 16);                     // tile_dim0 = n
    d1[4] = 1;                                       // tile_dim1=1, tile_dim2=0
    d1[5] = n;                                       // tensor_dim0_stride = n
    d1[6] = 0;
    d1[7] = 0;
    i32x4 d2; d2[0] = 1; d2[1] = 1; d2[2] = 0; d2[3] = 0;         // dim2=1, dim3=1
    i32x4 d3; d3[0] = 0; d3[1] = (1 << 16); d3[2] = 0; d3[3] = 0; // dim4=1
    i32x8 d4;
    d4[0] = 0; d4[1] = 0; d4[2] = 0; d4[3] = 0;
    d4[4] = 0; d4[5] = 0; d4[6] = 0; d4[7] = 0;      // trailing group: unused/zero
    __builtin_amdgcn_tensor_load_to_lds(d0, d1, d2, d3, d4, 0);
    __builtin_amdgcn_s_wait_tensorcnt(0);
  }
  __syncthreads();

  // ---- Phase 2: per-position blend weight + neighbor id --------------------
  for (int p = cb + (int)threadIdx.x; p < cb + CHUNK; p += NTHREADS) {
    const int id = sid[p - lo];
    int s = p;
    while (s > lo && sid[s - 1 - lo] == id) --s;     // halo >> max run length
    int e = p + 1;
    while (e < hi && sid[e - lo] == id) ++e;
    const int dur = e - s;

    int dur_prev = dur, pid = id;
    if (s > 0) {
      pid = sid[s - 1 - lo];
      int ps = s - 1;
      while (ps > lo && sid[ps - 1 - lo] == pid) --ps;
      dur_prev = s - ps;
    }
    int dur_next = dur, nid = id;
    if (e < T_LEN) {
      nid = sid[e - lo];
      int ne = e + 1;
      while (ne < hi && sid[ne - lo] == nid) ++ne;
      dur_next = ne - e;
    }

    // right edge of this segment: ramp toward next phoneme
    const float radn = 0.3f * (float)min(dur, dur_next);
    const int   rn   = max(1, (int)rintf(radn));     // round half-to-even
    const int   lst  = max(0, e - rn);
    float wn = 0.0f;
    if ((e < T_LEN) && (radn >= 0.5f) && (p >= lst))
      wn = fminf(0.5f * (float)(p - lst + 1) / (float)rn, 0.5f);

    // left edge: ramp toward previous phoneme
    const float radp = 0.3f * (float)min(dur_prev, dur);
    const int   rp   = max(1, (int)rintf(radp));
    const int   rnd  = min(T_LEN, s + rp);
    float wp = 0.0f;
    if ((s > 0) && (radp >= 0.5f) && (p < rnd))
      wp = fminf(0.5f * (float)(rnd - p) / (float)rp, 0.5f);

    sw [p - cb] = fmaxf(wp, wn);
    snb[p - cb] = (wn > wp) ? nid : ((wp > 0.0f) ? pid : id);
  }
  __syncthreads();

  // ---- Phase 3: gather from L2-resident table, blend, streamed store -------
  const f32x4* t4 = reinterpret_cast<const f32x4*>(table);
  f32x4* o4 = reinterpret_cast<f32x4*>(out);
  const size_t rowbase = ((size_t)b * T_LEN + cb) * (D_DIM / 4);
  for (int i = (int)threadIdx.x; i < CHUNK * (D_DIM / 4); i += NTHREADS) {
    const int p  = i >> 4;
    const int q  = i & 15;
    const int id = sid[off + p];
    const float w = sw[p];
    const int  nb = snb[p];
    f32x4 ev = {0.f, 0.f, 0.f, 0.f};
    f32x4 nv = {0.f, 0.f, 0.f, 0.f};
    if (id != 0) ev = t4[(size_t)id * (D_DIM / 4) + q];   // padding_idx=0 -> zeros
    if (nb != 0) nv = t4[(size_t)nb * (D_DIM / 4) + q];
    const float a = 1.0f - w;
    f32x4 o = a * ev + w * nv;                            // (1-w)*emb + w*nemb
    // output is write-once / never re-read: stream past the caches
    __builtin_nontemporal_store(o, &o4[rowbase + (size_t)p * (D_DIM / 4) + q]);
  }
}

extern "C" void kernel_launch(void* const* d_in, const int* in_sizes, int n_in,
                              void* d_out, int out_size, void* d_ws, size_t ws_size,
                              hipStream_t stream) {
  (void)n_in; (void)out_size; (void)d_ws; (void)ws_size;
  const int*   ids   = (const int*)d_in[0];     // (B, 8192) int
  const float* table = (const float*)d_in[1];   // (2820, 64) f32
  float*       out   = (float*)d_out;           // (B, 8192, 64) f32
  const int B = in_sizes[0] / T_LEN;            // 32
  dim3 grid(B * (T_LEN / CHUNK));               // 512 blocks
  blur_embed_kernel<<<grid, NTHREADS, 0, stream>>>(ids, table, out);
}